// CapsLinear_28750511079696
// MI455X (gfx1250) — compile-verified
//
#include <hip/hip_runtime.h>

// CapsLinear forward for MI455X (gfx1250, wave32).
//
// Shapes: x(64,1152,8) f32, W(64,1152,16,8) f32 -> out(64,64,16) f32.
//
// Roofline: x_hat is 302 MB; 1 write + ~5 streaming reads across 3 routing
// iterations ~= 1.8 GB HBM => ~78 us at 23.3 TB/s. Compute (1.2 GFLOP) is
// negligible, so kernels are shaped for streaming bandwidth:
//  - x_hat generated with V_WMMA_F32_16X16X4_F32 (full f32, 4 batch tiles of
//    M=16, 2 K-steps of 4, N=16=D_OUT), one wave per (o,i).
//  - x_hat stored [b][i][o][do] so softmax-over-o, weighted reduce and
//    agreement are all coalesced b128 streams.
//  - k_route keeps per-lane partial sums in registers (lane owns o and o+32),
//    flushing to LDS once per block, then one global atomic pass.

typedef __attribute__((ext_vector_type(2))) float v2f;
typedef __attribute__((ext_vector_type(4))) float v4f;
typedef __attribute__((ext_vector_type(8))) float v8f;

#define B_SZ  64
#define N_IN  1152
#define D_IN  8
#define N_OUT 64
#define D_OUT 16
#define EPS_F 1e-8f

// ---------------------------------------------------------------------------
// Kernel 1: x_hat[b][i][o][do] = sum_di W[o,i,do,di] * x[b,i,di]
// One wave per (o,i). grid = (N_IN/8, N_OUT), block = 256 (8 waves).
// A-tile (16x4 f32): lanes 0-15 hold M=0..15 with K={k0,k0+1}; lanes 16-31 K={k0+2,k0+3}.
// B-tile (4x16 f32): lanes 0-15 hold N=0..15 with K={k0,k0+1}; lanes 16-31 K={k0+2,k0+3}.
// D (16x16 f32):     VGPR r -> M = r + 8*(lane>=16), N = lane&15.
// ---------------------------------------------------------------------------
__global__ __launch_bounds__(256) void k_xhat(const float* __restrict__ x,
                                              const float* __restrict__ W,
                                              float* __restrict__ xh) {
  const int wave = threadIdx.x >> 5;
  const int lane = threadIdx.x & 31;
  const int i    = blockIdx.x * 8 + wave;
  const int o    = blockIdx.y;
  const int half = lane >> 4;     // which K-half this lane holds
  const int l    = lane & 15;     // M index (A) / N index (B,D)

  // B operand: B[k][n] = W[o, i, do=n, di=k]
  const float* wp = W + (((size_t)o * N_IN + i) * D_OUT + l) * D_IN + half * 2;
  v2f wb0 = { wp[0], wp[1] };     // K-step 0: di = half*2, half*2+1
  v2f wb1 = { wp[4], wp[5] };     // K-step 1: di = 4+half*2, 4+half*2+1

#pragma unroll
  for (int t = 0; t < 4; ++t) {   // 4 batch tiles of 16
    const float* xp = x + ((size_t)(t * 16 + l) * N_IN + i) * D_IN + half * 2;
    v2f a0 = { xp[0], xp[1] };
    v2f a1 = { xp[4], xp[5] };
    v8f c = {};
    c = __builtin_amdgcn_wmma_f32_16x16x4_f32(false, a0, false, wb0,
                                              (short)0, c, false, false);
    c = __builtin_amdgcn_wmma_f32_16x16x4_f32(false, a1, false, wb1,
                                              (short)0, c, false, false);
#pragma unroll
    for (int r = 0; r < 8; ++r) {
      const int bb = t * 16 + r + half * 8;          // batch index (M)
      xh[(((size_t)bb * N_IN + i) * N_OUT + o) * D_OUT + l] = c[r];
    }
  }
}

// ---------------------------------------------------------------------------
// Kernel 2: softmax over o (per b,i) + weighted partial sums into S[b][o][do].
// grid = (B, 16 chunks), block = 256 (8 waves). Each wave owns one i at a
// time: lane holds logits for o = lane and o = lane+32; wave32 shuffle
// reductions give the softmax. Weighted sums accumulate in REGISTERS
// (lane owns its o pair for the whole loop), flushed once per block through
// LDS (ds_add_f32) and once to global S (global_atomic_add_f32).
// ---------------------------------------------------------------------------
__global__ __launch_bounds__(256) void k_route(const float* __restrict__ xh,
                                               const float* __restrict__ blog,
                                               float* __restrict__ S) {
  __shared__ float sacc[N_OUT * D_OUT];      // 4 KB cross-wave merge buffer
  const int b = blockIdx.x;
  const int chunk = blockIdx.y;
  for (int t = threadIdx.x; t < N_OUT * D_OUT; t += 256) sacc[t] = 0.0f;
  __syncthreads();

  const int wave = threadIdx.x >> 5;
  const int lane = threadIdx.x & 31;
  const int I_PER = N_IN / 16;               // 72 input capsules per chunk
  const int i0 = chunk * I_PER;

  v4f acc[2][4];                             // [o-pair][quarter of D_OUT]
#pragma unroll
  for (int p = 0; p < 2; ++p)
#pragma unroll
    for (int q = 0; q < 4; ++q) acc[p][q] = (v4f){0.f, 0.f, 0.f, 0.f};

  for (int ii = wave; ii < I_PER; ii += 8) {
    const int i = i0 + ii;
    const float* lg = blog + ((size_t)b * N_IN + i) * N_OUT;
    float e0 = lg[lane];
    float e1 = lg[lane + 32];
    float m = fmaxf(e0, e1);
#pragma unroll
    for (int s = 16; s > 0; s >>= 1) m = fmaxf(m, __shfl_xor(m, s, 32));
    e0 = __expf(e0 - m);
    e1 = __expf(e1 - m);
    float sum = e0 + e1;
#pragma unroll
    for (int s = 16; s > 0; s >>= 1) sum += __shfl_xor(sum, s, 32);
    const float inv = __frcp_rn(sum);
    const float c01[2] = { e0 * inv, e1 * inv };

    const float* xb = xh + ((size_t)b * N_IN + i) * (N_OUT * D_OUT);
#pragma unroll
    for (int p = 0; p < 2; ++p) {
      const int o = lane + p * 32;
      const float cc = c01[p];
      const v4f* xo = (const v4f*)(xb + (size_t)o * D_OUT);
#pragma unroll
      for (int q = 0; q < 4; ++q) {
        v4f val = xo[q];
        acc[p][q].x = fmaf(cc, val.x, acc[p][q].x);
        acc[p][q].y = fmaf(cc, val.y, acc[p][q].y);
        acc[p][q].z = fmaf(cc, val.z, acc[p][q].z);
        acc[p][q].w = fmaf(cc, val.w, acc[p][q].w);
      }
    }
  }

  // One LDS flush per lane (waves collide on the same o rows -> ds_add_f32).
#pragma unroll
  for (int p = 0; p < 2; ++p) {
    const int o = lane + p * 32;
#pragma unroll
    for (int q = 0; q < 4; ++q) {
      atomicAdd(&sacc[o * D_OUT + q * 4 + 0], acc[p][q].x);
      atomicAdd(&sacc[o * D_OUT + q * 4 + 1], acc[p][q].y);
      atomicAdd(&sacc[o * D_OUT + q * 4 + 2], acc[p][q].z);
      atomicAdd(&sacc[o * D_OUT + q * 4 + 3], acc[p][q].w);
    }
  }
  __syncthreads();
  float* sg = S + (size_t)b * (N_OUT * D_OUT);
  for (int t = threadIdx.x; t < N_OUT * D_OUT; t += 256)
    atomicAdd(&sg[t], sacc[t]);
}

// ---------------------------------------------------------------------------
// Kernel 3: squash. One thread per (b,o): v = (n2/(1+n2)) * s * rsqrt(n2+eps).
// grid = B, block = N_OUT.
// ---------------------------------------------------------------------------
__global__ __launch_bounds__(64) void k_squash(const float* __restrict__ S,
                                               float* __restrict__ v) {
  const int b = blockIdx.x;
  const int o = threadIdx.x;
  const float* sp = S + ((size_t)b * N_OUT + o) * D_OUT;
  float vals[D_OUT];
  float n2 = 0.0f;
#pragma unroll
  for (int d = 0; d < D_OUT; ++d) {
    vals[d] = sp[d];
    n2 = fmaf(vals[d], vals[d], n2);
  }
  const float scale = (n2 / (1.0f + n2)) * rsqrtf(n2 + EPS_F);
  float* vp = v + ((size_t)b * N_OUT + o) * D_OUT;
#pragma unroll
  for (int d = 0; d < D_OUT; ++d) vp[d] = vals[d] * scale;
}

// ---------------------------------------------------------------------------
// Kernel 4: agreement. b_logits[b][i][o] += dot(v[b,o,:], x_hat[b,i,o,:]).
// One thread per (b,i,o); each thread reads its private 64 B of x_hat with
// b128 loads, v[b,o,:] comes from L2; logit writes fully coalesced.
// ---------------------------------------------------------------------------
__global__ __launch_bounds__(256) void k_agree(const float* __restrict__ xh,
                                               const float* __restrict__ v,
                                               float* __restrict__ blog) {
  const size_t idx = (size_t)blockIdx.x * 256 + threadIdx.x;  // (b*N_IN+i)*N_OUT+o
  const int o = (int)(idx & (N_OUT - 1));
  const int b = (int)(idx / ((size_t)N_IN * N_OUT));
  const v4f* xo = (const v4f*)(xh + idx * D_OUT);
  const v4f* vo = (const v4f*)(v + ((size_t)b * N_OUT + o) * D_OUT);
  float acc = 0.0f;
#pragma unroll
  for (int q = 0; q < 4; ++q) {
    v4f a = xo[q];
    v4f w = vo[q];
    acc = fmaf(a.x, w.x, acc);
    acc = fmaf(a.y, w.y, acc);
    acc = fmaf(a.z, w.z, acc);
    acc = fmaf(a.w, w.w, acc);
  }
  blog[idx] += acc;
}

// ---------------------------------------------------------------------------
extern "C" void kernel_launch(void* const* d_in, const int* in_sizes, int n_in,
                              void* d_out, int out_size, void* d_ws, size_t ws_size,
                              hipStream_t stream) {
  (void)in_sizes; (void)n_in; (void)out_size; (void)ws_size;
  const float* x = (const float*)d_in[0];   // (64,1152,8)
  const float* W = (const float*)d_in[1];   // (64,1152,16,8)
  float* out = (float*)d_out;               // (64,64,16)

  const size_t xh_elems   = (size_t)B_SZ * N_IN * N_OUT * D_OUT;  // 75.5M
  const size_t blog_elems = (size_t)B_SZ * N_IN * N_OUT;          // 4.72M
  const size_t s_elems    = (size_t)B_SZ * N_OUT * D_OUT;         // 65536

  float* xh   = (float*)d_ws;
  float* blog = xh + xh_elems;
  float* S    = blog + blog_elems;
  float* v    = S + s_elems;

  // b logits start at zero every call (deterministic).
  hipMemsetAsync(blog, 0, blog_elems * sizeof(float), stream);

  // x_hat via WMMA: one wave per (o,i).
  k_xhat<<<dim3(N_IN / 8, N_OUT), 256, 0, stream>>>(x, W, xh);

  for (int it = 0; it < 3; ++it) {
    hipMemsetAsync(S, 0, s_elems * sizeof(float), stream);
    k_route<<<dim3(B_SZ, 16), 256, 0, stream>>>(xh, blog, S);
    float* dst = (it == 2) ? out : v;
    k_squash<<<B_SZ, N_OUT, 0, stream>>>(S, dst);
    if (it < 2) {
      const int total = B_SZ * N_IN * N_OUT;   // 4,718,592
      k_agree<<<total / 256, 256, 0, stream>>>(xh, v, blog);
    }
  }
}